// RefineCostVolume_11192684773831
// MI455X (gfx1250) — compile-verified
//
#include <hip/hip_runtime.h>
#include <hip/hip_bf16.h>

typedef float v2f __attribute__((ext_vector_type(2)));
typedef float v8f __attribute__((ext_vector_type(8)));

// Problem constants (from reference setup_inputs)
#define IMG_H 96
#define IMG_W 192
#define IMG_L (IMG_H * IMG_W)   // 18432
#define NCH 32
#define NHEADS 4
#define HDIM 8
#define PWIN 121
#define RGN 26                   // 16 + 2*5 halo
#define RGN2 (RGN * RGN)         // 676

// ---------------------------------------------------------------------------
// Kernel 1: fused mask head. 8x8 output tile per block, halo-3 input staged
// in LDS, conv1+BN+ReLU -> conv2+ReLU -> conv3+sigmoid, buffers aliased.
// Grid: 4 (img,b) * 12 * 24 tiles = 1152 blocks of 256 threads.
// ---------------------------------------------------------------------------
__global__ __launch_bounds__(256) void mask_head_kernel(
    const float* __restrict__ featL, const float* __restrict__ featR,
    const float* __restrict__ w1, const float* __restrict__ bng,
    const float* __restrict__ bnb, const float* __restrict__ bnm,
    const float* __restrict__ bnv, const float* __restrict__ w2,
    const float* __restrict__ w3, float* __restrict__ mask_out,
    float* __restrict__ wbin)
{
    __shared__ float smem[14 * 14 * 32 + 12 * 12 * 32]; // 43.5 KB
    float* s_in = smem;                 // 14x14x32 input stage (halo 3)
    float* s_t1 = smem + 14 * 14 * 32;  // 12x12x32 conv1 output (halo 2)
    float* s_t2 = smem;                 // 10x10x32 conv2 output aliases s_in

    int blk  = blockIdx.x;
    int imgb = blk / 288;               // (side<<1)|batch
    int t    = blk % 288;
    int ty0  = (t / 24) * 8;
    int tx0  = (t % 24) * 8;
    const float* feat = ((imgb >> 1) ? featR : featL) + (size_t)(imgb & 1) * NCH * IMG_L;
    int tid = threadIdx.x;

    // stage input, zero-padded (layout [pixel][ic])
    for (int i = tid; i < 14 * 14 * 32; i += 256) {
        int ic = i & 31, pix = i >> 5;
        int py = pix / 14, px = pix % 14;
        int yy = ty0 + py - 3, xx = tx0 + px - 3;
        float v = 0.f;
        if (yy >= 0 && yy < IMG_H && xx >= 0 && xx < IMG_W)
            v = feat[(size_t)ic * IMG_L + yy * IMG_W + xx];
        s_in[pix * 32 + ic] = v;
    }
    __syncthreads();

    // conv1 + BN + ReLU -> t1 (12x12)
    for (int i = tid; i < 12 * 12 * 32; i += 256) {
        int oc = i & 31, pix = i >> 5;
        int py = pix / 12, px = pix % 12;
        float acc = 0.f;
        for (int ky = 0; ky < 3; ++ky)
            for (int kx = 0; kx < 3; ++kx) {
                const float* ip = &s_in[((py + ky) * 14 + px + kx) * 32];
                const float* wp = &w1[oc * 288 + ky * 3 + kx];
#pragma unroll
                for (int ic = 0; ic < 32; ++ic) acc = fmaf(ip[ic], wp[ic * 9], acc);
            }
        float y = bng[oc] * (acc - bnm[oc]) * rsqrtf(bnv[oc] + 1e-5f) + bnb[oc];
        s_t1[pix * 32 + oc] = fmaxf(y, 0.f);
    }
    __syncthreads();

    // conv2 + ReLU -> t2 (10x10), reusing the dead input buffer
    for (int i = tid; i < 10 * 10 * 32; i += 256) {
        int oc = i & 31, pix = i >> 5;
        int py = pix / 10, px = pix % 10;
        float acc = 0.f;
        for (int ky = 0; ky < 3; ++ky)
            for (int kx = 0; kx < 3; ++kx) {
                const float* ip = &s_t1[((py + ky) * 12 + px + kx) * 32];
                const float* wp = &w2[oc * 288 + ky * 3 + kx];
#pragma unroll
                for (int ic = 0; ic < 32; ++ic) acc = fmaf(ip[ic], wp[ic * 9], acc);
            }
        s_t2[pix * 32 + oc] = fmaxf(acc, 0.f);
    }
    __syncthreads();

    // conv3 + sigmoid (8x8, single output channel)
    if (tid < 64) {
        int py = tid / 8, px = tid % 8;
        float acc = 0.f;
        for (int ky = 0; ky < 3; ++ky)
            for (int kx = 0; kx < 3; ++kx) {
                const float* ip = &s_t2[((py + ky) * 10 + px + kx) * 32];
                const float* wp = &w3[ky * 3 + kx];
#pragma unroll
                for (int ic = 0; ic < 32; ++ic) acc = fmaf(ip[ic], wp[ic * 9], acc);
            }
        float sg = 1.f / (1.f + __expf(-acc));
        int yy = ty0 + py, xx = tx0 + px;
        int img = imgb >> 1, b = imgb & 1;
        mask_out[(size_t)img * 2 * IMG_L + (size_t)b * IMG_L + yy * IMG_W + xx] = sg;
        wbin[(size_t)imgb * IMG_L + yy * IMG_W + xx] = (sg < 0.5f) ? 1.f : 0.f;
    }
}

// ---------------------------------------------------------------------------
// Kernel 2: qkv projection GEMM, Out(96 x L) = W(96x32) * X(32 x L) per image.
// One wave per 16x16 output tile; 8 chained V_WMMA_F32_16X16X4_F32 over K=32.
// ---------------------------------------------------------------------------
__global__ __launch_bounds__(256) void qkv_gemm_kernel(
    const float* __restrict__ featL, const float* __restrict__ featR,
    const float* __restrict__ qkv_w, float* __restrict__ qkv_out)
{
    const int L = IMG_L;
    const int MT = 6, NT = L / 16; // 6 x 1152 tiles per (img,b)
    int wave = (blockIdx.x * 256 + threadIdx.x) >> 5;
    int lane = threadIdx.x & 31;
    int imgb = wave / (MT * NT);
    if (imgb >= 4) return;
    int t  = wave % (MT * NT);
    int mT = t / NT, nT = t % NT;

    const float* feat = ((imgb >> 1) ? featR : featL) + (size_t)(imgb & 1) * NCH * L;
    float* outp = qkv_out + (size_t)imgb * 96 * L;

    int mrow = mT * 16 + (lane & 15);
    int n    = nT * 16 + (lane & 15);
    int kB   = (lane >> 4) * 2;

    v8f acc = {0.f, 0.f, 0.f, 0.f, 0.f, 0.f, 0.f, 0.f};
#pragma unroll
    for (int kk = 0; kk < 32; kk += 4) {
        v2f a, b;
        a.x = qkv_w[mrow * 32 + kk + kB + 0];
        a.y = qkv_w[mrow * 32 + kk + kB + 1];
        b.x = feat[(size_t)(kk + kB + 0) * L + n];
        b.y = feat[(size_t)(kk + kB + 1) * L + n];
        acc = __builtin_amdgcn_wmma_f32_16x16x4_f32(false, a, false, b,
                                                    (short)0, acc, false, false);
    }
    int mBase = mT * 16 + ((lane >> 4) ? 8 : 0);
#pragma unroll
    for (int r = 0; r < 8; ++r)
        outp[(size_t)(mBase + r) * L + n] = acc[r];
}

// ---------------------------------------------------------------------------
// Kernel 3: 11x11 masked window attention, online softmax. 16x16 pixel tile
// per block (1 pixel/thread). k/v halo (26x26 per head) staged directly into
// LDS with GLOBAL_LOAD_ASYNC_TO_LDS_B32 (ASYNCcnt path, no VGPR round trip);
// out-of-image halo lanes write zeros (matches unfold's zero padding).
// LDS layout is [d][pix]: coalesced on the global side, stride-1 (bank-
// conflict-free) on the compute side.
// ---------------------------------------------------------------------------
__global__ __launch_bounds__(256) void attn_kernel(
    const float* __restrict__ qkv, const float* __restrict__ wbin,
    const float* __restrict__ pos_bias, float* __restrict__ attn_out)
{
    const int L = IMG_L;
    __shared__ float s_w[RGN2];
    __shared__ float s_pb[NHEADS * PWIN];
    __shared__ float s_k[HDIM * RGN2];
    __shared__ float s_v[HDIM * RGN2];

    int blk  = blockIdx.x;
    int imgb = blk / 72;
    int t    = blk % 72;               // 6 x 12 tiles of 16x16
    int ty0  = (t / 12) * 16;
    int tx0  = (t % 12) * 16;
    const float* qkvB = qkv + (size_t)imgb * 96 * L;
    const float* wB   = wbin + (size_t)imgb * L;
    float* outB       = attn_out + (size_t)imgb * NCH * L;

    int tid = threadIdx.x;
    int px = tid & 15, py = tid >> 4;
    int gy = ty0 + py, gx = tx0 + px;
    int l = gy * IMG_W + gx;

    for (int i = tid; i < NHEADS * PWIN; i += 256) s_pb[i] = pos_bias[i];

    // stage binary window weights (zero outside image = unfold padding)
    for (int i = tid; i < RGN2; i += 256) {
        int ry = i / RGN, rx = i % RGN;
        int yy = ty0 + ry - 5, xx = tx0 + rx - 5;
        float w = 0.f;
        if (yy >= 0 && yy < IMG_H && xx >= 0 && xx < IMG_W) w = wB[yy * IMG_W + xx];
        s_w[i] = w;
    }

    const float scale = 0.3535533905932738f; // 8^-0.5

    for (int h = 0; h < NHEADS; ++h) {
        __syncthreads();
        // async-stage k, v halo for this head: memory -> LDS directly
        const float* kRow = qkvB + (size_t)(32 + h * HDIM) * L;
        const float* vRow = qkvB + (size_t)(64 + h * HDIM) * L;
#pragma unroll
        for (int d = 0; d < HDIM; ++d) {
            for (int pix = tid; pix < RGN2; pix += 256) {
                int ry = pix / RGN, rx = pix % RGN;
                int yy = ty0 + ry - 5, xx = tx0 + rx - 5;
                float* kDst = &s_k[d * RGN2 + pix];
                float* vDst = &s_v[d * RGN2 + pix];
                if (yy >= 0 && yy < IMG_H && xx >= 0 && xx < IMG_W) {
                    int ll = yy * IMG_W + xx;
                    const float* gk = kRow + (size_t)d * L + ll;
                    const float* gv = vRow + (size_t)d * L + ll;
                    unsigned lk = (unsigned)(uintptr_t)kDst; // LDS byte offset
                    unsigned lv = (unsigned)(uintptr_t)vDst;
                    asm volatile(
                        "global_load_async_to_lds_b32 %0, %1, off\n\t"
                        "global_load_async_to_lds_b32 %2, %3, off"
                        :: "v"(lk), "v"(gk), "v"(lv), "v"(gv)
                        : "memory");
                } else {
                    *kDst = 0.f;
                    *vDst = 0.f;
                }
            }
        }
#if __has_builtin(__builtin_amdgcn_s_wait_asynccnt)
        __builtin_amdgcn_s_wait_asynccnt(0);
#else
        asm volatile("s_wait_asynccnt 0x0" ::: "memory");
#endif
        __syncthreads();

        float q[HDIM];
#pragma unroll
        for (int d = 0; d < HDIM; ++d) q[d] = qkvB[(size_t)(h * HDIM + d) * L + l];

        float m = -3.0e38f, s = 0.f;
        float o[HDIM] = {0.f, 0.f, 0.f, 0.f, 0.f, 0.f, 0.f, 0.f};
        for (int p = 0; p < PWIN; ++p) {
            int dy = p / 11 - 5, dx = p % 11 - 5;
            int rIdx = (py + 5 + dy) * RGN + (px + 5 + dx);
            float wgt = s_w[rIdx];
            float dot = 0.f;
#pragma unroll
            for (int d = 0; d < HDIM; ++d) dot = fmaf(q[d], s_k[d * RGN2 + rIdx], dot);
            float logit = scale * wgt * dot + s_pb[h * PWIN + p];
            float nm = fmaxf(m, logit);
            float f = __expf(m - nm);
            float e = __expf(logit - nm);
            s = s * f + e;
            float ew = e * wgt;
#pragma unroll
            for (int d = 0; d < HDIM; ++d) o[d] = fmaf(o[d], f, ew * s_v[d * RGN2 + rIdx]);
            m = nm;
        }
        float inv = 1.f / s;
#pragma unroll
        for (int d = 0; d < HDIM; ++d)
            outB[(size_t)(h * HDIM + d) * L + l] = o[d] * inv;
    }
}

// ---------------------------------------------------------------------------
// Kernel 4: output projection GEMM (32x32 x L) fused with bias + residual.
// ---------------------------------------------------------------------------
__global__ __launch_bounds__(256) void proj_gemm_kernel(
    const float* __restrict__ featL, const float* __restrict__ featR,
    const float* __restrict__ attn_out, const float* __restrict__ proj_w,
    const float* __restrict__ proj_b, float* __restrict__ out)
{
    const int L = IMG_L;
    const int MT = 2, NT = L / 16;
    int wave = (blockIdx.x * 256 + threadIdx.x) >> 5;
    int lane = threadIdx.x & 31;
    int imgb = wave / (MT * NT);
    if (imgb >= 4) return;
    int t  = wave % (MT * NT);
    int mT = t / NT, nT = t % NT;

    const float* feat = ((imgb >> 1) ? featR : featL) + (size_t)(imgb & 1) * NCH * L;
    const float* X    = attn_out + (size_t)imgb * NCH * L;
    float* o = out + (size_t)(imgb >> 1) * (2 * NCH * L) + (size_t)(imgb & 1) * NCH * L;

    int mrow = mT * 16 + (lane & 15);
    int n    = nT * 16 + (lane & 15);
    int kB   = (lane >> 4) * 2;

    v8f acc = {0.f, 0.f, 0.f, 0.f, 0.f, 0.f, 0.f, 0.f};
#pragma unroll
    for (int kk = 0; kk < 32; kk += 4) {
        v2f a, b;
        a.x = proj_w[mrow * 32 + kk + kB + 0];
        a.y = proj_w[mrow * 32 + kk + kB + 1];
        b.x = X[(size_t)(kk + kB + 0) * L + n];
        b.y = X[(size_t)(kk + kB + 1) * L + n];
        acc = __builtin_amdgcn_wmma_f32_16x16x4_f32(false, a, false, b,
                                                    (short)0, acc, false, false);
    }
    int mBase = mT * 16 + ((lane >> 4) ? 8 : 0);
#pragma unroll
    for (int r = 0; r < 8; ++r) {
        int mm = mBase + r;
        o[(size_t)mm * L + n] = acc[r] + proj_b[mm] + feat[(size_t)mm * L + n];
    }
}

// ---------------------------------------------------------------------------
extern "C" void kernel_launch(void* const* d_in, const int* in_sizes, int n_in,
                              void* d_out, int out_size, void* d_ws, size_t ws_size,
                              hipStream_t stream)
{
    const float* featL  = (const float*)d_in[0];
    const float* featR  = (const float*)d_in[1];
    const float* mh_w1  = (const float*)d_in[2];
    const float* bng    = (const float*)d_in[3];
    const float* bnb    = (const float*)d_in[4];
    const float* bnm    = (const float*)d_in[5];
    const float* bnv    = (const float*)d_in[6];
    const float* mh_w2  = (const float*)d_in[7];
    const float* mh_w3  = (const float*)d_in[8];
    const float* qkv_w  = (const float*)d_in[9];
    const float* proj_w = (const float*)d_in[10];
    const float* proj_b = (const float*)d_in[11];
    const float* posb   = (const float*)d_in[12];

    float* out = (float*)d_out;
    float* ws  = (float*)d_ws;

    // workspace layout (floats)
    const size_t qkvSize  = 4ull * 96 * IMG_L;   // 28.3 MB
    const size_t attnSize = 4ull * NCH * IMG_L;  // 9.4 MB
    const size_t wbinSize = 4ull * IMG_L;        // 0.3 MB
    if (ws_size < (qkvSize + attnSize + wbinSize) * sizeof(float)) return;
    float* qkv   = ws;
    float* attnO = ws + qkvSize;
    float* wbin  = ws + qkvSize + attnSize;

    const size_t perFeat = 2ull * NCH * IMG_L;   // 1,179,648 floats
    float* maskOut = out + 2 * perFeat;          // masks follow featL_ref, featR_ref

    mask_head_kernel<<<1152, 256, 0, stream>>>(featL, featR, mh_w1, bng, bnb, bnm,
                                               bnv, mh_w2, mh_w3, maskOut, wbin);
    qkv_gemm_kernel<<<3456, 256, 0, stream>>>(featL, featR, qkv_w, qkv);
    attn_kernel<<<288, 256, 0, stream>>>(qkv, wbin, posb, attnO);
    proj_gemm_kernel<<<1152, 256, 0, stream>>>(featL, featR, attnO, proj_w, proj_b, out);
}